// NAtt_6098853560422
// MI455X (gfx1250) — compile-verified
//
#include <hip/hip_runtime.h>
#include <hip/hip_bf16.h>
#include <cstdint>

// ---------------- WMMA types (CDNA5 / gfx1250, wave32) ----------------
typedef __attribute__((ext_vector_type(2))) float v2f;
typedef __attribute__((ext_vector_type(8))) float v8f;

#define NEG_SLOPE 0.2f

// Monotonic float<->uint encoding for atomicMax-based segment max.
__device__ __forceinline__ unsigned fenc(float f) {
    unsigned u = __float_as_uint(f);
    return (u & 0x80000000u) ? ~u : (u | 0x80000000u);
}
__device__ __forceinline__ float fdec(unsigned u) {
    return (u & 0x80000000u) ? __uint_as_float(u ^ 0x80000000u) : __uint_as_float(~u);
}
#define ENC_NEG_INF 0x007FFFFFu   // fenc(-inf)

// ---------------- Fill kernels ----------------
__global__ void fill_f32(float* p, float v, size_t n) {
    size_t i = (size_t)blockIdx.x * blockDim.x + threadIdx.x;
    size_t stride = (size_t)gridDim.x * blockDim.x;
    for (; i < n; i += stride) p[i] = v;
}
__global__ void fill_u32(unsigned* p, unsigned v, size_t n) {
    size_t i = (size_t)blockIdx.x * blockDim.x + threadIdx.x;
    size_t stride = (size_t)gridDim.x * blockDim.x;
    for (; i < n; i += stride) p[i] = v;
}

// ---------------- Tiled transpose: Bt[Nc,K] = B[K,Nc]^T ----------------
// block = 256 (32x8), grid = (ceil(K/32), ceil(Nc/32))
__global__ void transpose_f32(const float* __restrict__ B, float* __restrict__ Bt,
                              int K, int Nc) {
    __shared__ float tile[32][33];
    const int kb = blockIdx.x * 32, nb = blockIdx.y * 32;
    const int tx = threadIdx.x & 31, ty = threadIdx.x >> 5;
#pragma unroll
    for (int j = ty; j < 32; j += 8) {
        int k = kb + j, n = nb + tx;
        tile[j][tx] = (k < K && n < Nc) ? B[(size_t)k * Nc + n] : 0.0f;
    }
    __syncthreads();
#pragma unroll
    for (int j = ty; j < 32; j += 8) {
        int n = nb + j, k = kb + tx;
        if (n < Nc && k < K) Bt[(size_t)n * K + k] = tile[tx][j];
    }
}

// ---------------- WMMA f32 GEMM: D[M,Nc] = A[M,K] @ Bt[Nc,K]^T ----------------
// Requirements: M % 16 == 0, K % 4 == 0 (software-pipelined by one K-step).
// Out-of-range columns clamped for addressing (column-independent WMMA) and
// masked only at the store, so the inner loop is branch-free with EXEC all-ones.
// grid = (M/16, ceil(Nc/(32*8))), block = 256 (8 waves). Wave -> two 16x16 tiles.
__global__ void gemm_f32_wmma(const float* __restrict__ A, const float* __restrict__ Bt,
                              float* __restrict__ D, int M, int K, int Nc) {
    const int wave = threadIdx.x >> 5;
    const int lane = threadIdx.x & 31;
    const int l15  = lane & 15;
    const int hi   = lane >> 4;                    // 0: K pair {0,1}, 1: K pair {2,3}
    const int row0 = blockIdx.x * 16;
    const int colBase = (blockIdx.y * 8 + wave) * 32;
    if (colBase >= Nc) return;                     // wave-uniform early out
    const int col0 = colBase + l15;
    const int col1 = colBase + 16 + l15;
    const int c0 = col0 < Nc ? col0 : Nc - 1;      // clamp for addressing
    const int c1 = col1 < Nc ? col1 : Nc - 1;

    const float* __restrict__ ap  = A  + (size_t)(row0 + l15) * K + 2 * hi;
    const float* __restrict__ bp0 = Bt + (size_t)c0 * K + 2 * hi;
    const float* __restrict__ bp1 = Bt + (size_t)c1 * K + 2 * hi;

    v8f acc0 = {}, acc1 = {};
    // one-stage software pipeline: loads for step k+4 issue before WMMA of step k
    v2f a  = *(const v2f*)(ap);
    v2f b0 = *(const v2f*)(bp0);
    v2f b1 = *(const v2f*)(bp1);
#pragma unroll 4
    for (int k = 4; k < K; k += 4) {
        v2f an  = *(const v2f*)(ap + k);           // A frag: 16x4 f32 ISA layout
        v2f b0n = *(const v2f*)(bp0 + k);          // B frag: 4x16 f32 ISA layout
        v2f b1n = *(const v2f*)(bp1 + k);
        acc0 = __builtin_amdgcn_wmma_f32_16x16x4_f32(false, a, false, b0,
                                                     (short)0, acc0, false, false);
        acc1 = __builtin_amdgcn_wmma_f32_16x16x4_f32(false, a, false, b1,
                                                     (short)0, acc1, false, false);
        a = an; b0 = b0n; b1 = b1n;
    }
    acc0 = __builtin_amdgcn_wmma_f32_16x16x4_f32(false, a, false, b0,
                                                 (short)0, acc0, false, false);
    acc1 = __builtin_amdgcn_wmma_f32_16x16x4_f32(false, a, false, b1,
                                                 (short)0, acc1, false, false);

    // D layout: VGPR r, lanes 0-15 -> M=r; lanes 16-31 -> M=8+r
    if (col0 < Nc) {
        float* drow = D + (size_t)(row0 + hi * 8) * Nc + col0;
#pragma unroll
        for (int r = 0; r < 8; ++r) drow[(size_t)r * Nc] = acc0[r];
    }
    if (col1 < Nc) {
        float* drow = D + (size_t)(row0 + hi * 8) * Nc + col1;
#pragma unroll
        for (int r = 0; r < 8; ++r) drow[(size_t)r * Nc] = acc1[r];
    }
}

// ---------------- Per-node double dot: o1[n]=H[n,:].v1, o2[n]=H[n,:].v2 ----------------
__global__ void row_dot2(const float* __restrict__ Hm, const float* __restrict__ v1,
                         const float* __restrict__ v2, float* __restrict__ o1,
                         float* __restrict__ o2, int F) {
    const int n = blockIdx.x;
    const float* row = Hm + (size_t)n * F;
    float s1 = 0.f, s2 = 0.f;
    for (int f = threadIdx.x; f < F; f += blockDim.x) {
        float h = row[f];
        s1 += h * v1[f];
        s2 += h * v2[f];
    }
    __shared__ float r1[256];
    __shared__ float r2[256];
    r1[threadIdx.x] = s1; r2[threadIdx.x] = s2;
    __syncthreads();
    for (int s = blockDim.x >> 1; s > 0; s >>= 1) {
        if ((int)threadIdx.x < s) {
            r1[threadIdx.x] += r1[threadIdx.x + s];
            r2[threadIdx.x] += r2[threadIdx.x + s];
        }
        __syncthreads();
    }
    if (threadIdx.x == 0) { o1[n] = r1[0]; o2[n] = r2[0]; }
}

// ---------------- Edge pass 1: leaky_relu(e) + segment max via atomicMax ----------------
__global__ void edge_max(const int* __restrict__ ei, int E, int N,
                         const float* __restrict__ as, const float* __restrict__ ad,
                         float* __restrict__ ebuf, unsigned* __restrict__ menc) {
    int i = blockIdx.x * blockDim.x + threadIdx.x;
    int tot = E + N;
    if (i >= tot) return;
    int s = (i < E) ? ei[i]     : (i - E);   // self loops appended
    int d = (i < E) ? ei[E + i] : (i - E);
    float e = as[s] + ad[d];
    e = (e > 0.f) ? e : NEG_SLOPE * e;
    ebuf[i] = e;
    atomicMax(&menc[d], fenc(e));
}

// ---------------- Edge pass 2: exp(e - m[dst]) + segment sum ----------------
__global__ void edge_exp(const int* __restrict__ ei, int E, int N,
                         const float* __restrict__ ebuf, const unsigned* __restrict__ menc,
                         float* __restrict__ exbuf, float* __restrict__ ssum) {
    int i = blockIdx.x * blockDim.x + threadIdx.x;
    int tot = E + N;
    if (i >= tot) return;
    int d = (i < E) ? ei[E + i] : (i - E);
    float ex = __expf(ebuf[i] - fdec(menc[d]));
    exbuf[i] = ex;
    atomicAdd(&ssum[d], ex);
}

// ---------------- Edge aggregate (wide features): one block per edge ----------------
__global__ void edge_agg_big(const int* __restrict__ ei, int E, int N,
                             const float* __restrict__ exbuf, const float* __restrict__ ssum,
                             const float* __restrict__ Hsrc, float* __restrict__ Out, int F) {
    int i = blockIdx.x;
    int s = (i < E) ? ei[i]     : (i - E);
    int d = (i < E) ? ei[E + i] : (i - E);
    float alpha = exbuf[i] / ssum[d];
    const float* hs = Hsrc + (size_t)s * F;
    float* od = Out + (size_t)d * F;
    for (int f = threadIdx.x; f < F; f += blockDim.x)
        atomicAdd(&od[f], hs[f] * alpha);
}

// ---------------- Edge aggregate (narrow features): one thread per edge ----------------
__global__ void edge_agg_small(const int* __restrict__ ei, int E, int N,
                               const float* __restrict__ exbuf, const float* __restrict__ ssum,
                               const float* __restrict__ Hsrc, float* __restrict__ Out, int F) {
    int i = blockIdx.x * blockDim.x + threadIdx.x;
    int tot = E + N;
    if (i >= tot) return;
    int s = (i < E) ? ei[i]     : (i - E);
    int d = (i < E) ? ei[E + i] : (i - E);
    float alpha = exbuf[i] / ssum[d];
    const float* hs = Hsrc + (size_t)s * F;
    float* od = Out + (size_t)d * F;
    for (int c = 0; c < F; ++c) atomicAdd(&od[c], hs[c] * alpha);
}

// ---------------- bias + ReLU ----------------
__global__ void bias_relu(float* __restrict__ X, const float* __restrict__ b,
                          int F, size_t total) {
    size_t i = (size_t)blockIdx.x * blockDim.x + threadIdx.x;
    size_t stride = (size_t)gridDim.x * blockDim.x;
    for (; i < total; i += stride) {
        float v = X[i] + b[i % F];
        X[i] = v > 0.f ? v : 0.f;
    }
}

// ---------------- bias + log_softmax over C classes, one thread per node ----------------
__global__ void bias_log_softmax(const float* __restrict__ Z, const float* __restrict__ b,
                                 float* __restrict__ Out, int N, int C) {
    int n = blockIdx.x * blockDim.x + threadIdx.x;
    if (n >= N) return;
    float z[16];
    float mx = -3.402823466e38f;
    for (int c = 0; c < C; ++c) {
        z[c] = Z[(size_t)n * C + c] + b[c];
        mx = fmaxf(mx, z[c]);
    }
    float sum = 0.f;
    for (int c = 0; c < C; ++c) sum += __expf(z[c] - mx);
    float lse = mx + __logf(sum);
    for (int c = 0; c < C; ++c) Out[(size_t)n * C + c] = z[c] - lse;
}

// ---------------- Launch ----------------
extern "C" void kernel_launch(void* const* d_in, const int* in_sizes, int n_in,
                              void* d_out, int out_size, void* d_ws, size_t ws_size,
                              hipStream_t stream) {
    (void)n_in; (void)out_size; (void)ws_size;

    const float* x      = (const float*)d_in[0];
    const int*   ei     = (const int*)d_in[1];   // [2, E] int32 (JAX x64 disabled)
    const float* W1     = (const float*)d_in[2];
    const float* a1_src = (const float*)d_in[3];
    const float* a1_dst = (const float*)d_in[4];
    const float* b1     = (const float*)d_in[5];
    const float* W2     = (const float*)d_in[6];
    const float* a2_src = (const float*)d_in[7];
    const float* a2_dst = (const float*)d_in[8];
    const float* b2     = (const float*)d_in[9];
    float* out = (float*)d_out;

    const int H   = in_sizes[3];            // 4096
    const int C   = in_sizes[7];            // 10
    const int FIN = in_sizes[2] / H;        // 512
    const int N   = in_sizes[0] / FIN;      // 10000
    const int E   = in_sizes[1] / 2;        // 80000
    const int TOT = E + N;                  // edges + self loops

    // workspace layout (fp32 unless noted)
    const size_t NH = (size_t)N * H;
    float*    h1    = (float*)d_ws;                 // [N,H]  layer-1 linear out
    float*    hr    = h1 + NH;                      // [N,H]  layer-1 aggregated (then ReLU'd)
    float*    as1   = hr + NH;                      // [N]
    float*    ad1   = as1 + N;                      // [N]
    unsigned* menc  = (unsigned*)(ad1 + N);         // [N]  encoded segment max
    float*    ssum  = (float*)(menc + N);           // [N]  segment sum
    float*    ebuf  = ssum + N;                     // [E+N]
    float*    exbuf = ebuf + TOT;                   // [E+N]
    float*    Bt    = exbuf + TOT;                  // [max(FIN*H, H*C)] transposed weights
    // layer-2 buffers overlay h1 (free after layer-1 aggregation)
    float*    h2    = h1;                           // [N,C]
    float*    o2    = h1 + (size_t)N * C;           // [N,C]
    float*    as2   = o2 + (size_t)N * C;           // [N]
    float*    ad2   = as2 + N;                      // [N]

    const int B256 = 256;
    const int edgeBlocks = (TOT + B256 - 1) / B256;

    // ---- Layer 1 ----
    {   // Bt = W1^T  [H, FIN]
        dim3 tg((FIN + 31) / 32, (H + 31) / 32);
        transpose_f32<<<tg, B256, 0, stream>>>(W1, Bt, FIN, H);
        // h1 = x @ W1   (WMMA fp32). M=N (mult of 16), K=FIN
        dim3 grid(N / 16, (H + 255) / 256);
        gemm_f32_wmma<<<grid, B256, 0, stream>>>(x, Bt, h1, N, FIN, H);
    }
    row_dot2<<<N, B256, 0, stream>>>(h1, a1_src, a1_dst, as1, ad1, H);

    fill_u32<<<64, B256, 0, stream>>>(menc, ENC_NEG_INF, (size_t)N);
    fill_f32<<<64, B256, 0, stream>>>(ssum, 0.f, (size_t)N);
    fill_f32<<<2048, B256, 0, stream>>>(hr, 0.f, NH);

    edge_max<<<edgeBlocks, B256, 0, stream>>>(ei, E, N, as1, ad1, ebuf, menc);
    edge_exp<<<edgeBlocks, B256, 0, stream>>>(ei, E, N, ebuf, menc, exbuf, ssum);
    edge_agg_big<<<TOT, B256, 0, stream>>>(ei, E, N, exbuf, ssum, h1, hr, H);
    bias_relu<<<2048, B256, 0, stream>>>(hr, b1, H, NH);

    // ---- Layer 2 ----
    {   // Bt = W2^T  [C, H]
        dim3 tg((H + 31) / 32, (C + 31) / 32);
        transpose_f32<<<tg, B256, 0, stream>>>(W2, Bt, H, C);
        // h2 = hr @ W2 (Nc=C=10, clamped columns)
        dim3 grid(N / 16, 1);
        gemm_f32_wmma<<<grid, B256, 0, stream>>>(hr, Bt, h2, N, H, C);
    }
    row_dot2<<<N, 32, 0, stream>>>(h2, a2_src, a2_dst, as2, ad2, C);

    fill_u32<<<64, B256, 0, stream>>>(menc, ENC_NEG_INF, (size_t)N);
    fill_f32<<<64, B256, 0, stream>>>(ssum, 0.f, (size_t)N);
    fill_f32<<<64, B256, 0, stream>>>(o2, 0.f, (size_t)N * C);

    edge_max<<<edgeBlocks, B256, 0, stream>>>(ei, E, N, as2, ad2, ebuf, menc);
    edge_exp<<<edgeBlocks, B256, 0, stream>>>(ei, E, N, ebuf, menc, exbuf, ssum);
    edge_agg_small<<<edgeBlocks, B256, 0, stream>>>(ei, E, N, exbuf, ssum, h2, o2, C);

    bias_log_softmax<<<(N + B256 - 1) / B256, B256, 0, stream>>>(o2, b2, out, N, C);
}